// MakeSpectrogram_37649683317194
// MI455X (gfx1250) — compile-verified
//
#include <hip/hip_runtime.h>

// CDNA5 / gfx1250 spectrogram: f32 WMMA (V_WMMA_F32_16X16X4_F32) GEMM
// mag/phase epilogue. Wave32, WGP-resident weight tiles in padded LDS.

typedef __attribute__((ext_vector_type(2))) float v2f;
typedef __attribute__((ext_vector_type(4))) float v4f;
typedef __attribute__((ext_vector_type(8))) float v8f;

#define FFT_N     1024
#define HOP       256
#define NBATCH    32
#define XLEN      160000
#define NFRAMES   622          // (160000-1024)/256 + 1
#define NMAG      513          // FFT_N/2 + 1
#define KTILES    64           // 1024 / 16 frequency tiles
#define TTILES    39           // ceil(622/16) frame tiles
#define LDS_STRIDE 1028        // 1024 + 4 pad: bank-conflict-free, 16B aligned
#define WAVES     8
#define TPW       5            // ceil(TTILES / WAVES)

__global__ __launch_bounds__(256)
void MakeSpectrogram_wmma_f32(const float* __restrict__ x,
                              const float* __restrict__ rw,
                              const float* __restrict__ iw,
                              float* __restrict__ out)
{
    extern __shared__ float lds[];   // [2][16][LDS_STRIDE] floats = 131584 B

    const int kt  = blockIdx.x;      // frequency tile (0..63)
    const int b   = blockIdx.y;      // batch (0..31)
    const int tid = threadIdx.x;
    const int k0  = kt * 16;

    // ---- Stage 16 rows of each weight matrix into LDS (padded stride). ----
    // 2 matrices * 16 rows * 256 float4 = 8192 float4 copies, 32 per thread.
    for (int e = tid; e < 2 * 16 * 256; e += 256) {
        const int mat = e >> 12;            // 0 = real, 1 = imag
        const int rem = e & 4095;
        const int row = rem >> 8;           // 0..15
        const int c4  = rem & 255;          // float4 column
        const float* src = (mat ? iw : rw) + (size_t)(k0 + row) * FFT_N + c4 * 4;
        v4f v = *(const v4f*)src;
        *(v4f*)&lds[mat * (16 * LDS_STRIDE) + row * LDS_STRIDE + c4 * 4] = v;
    }
    __syncthreads();

    const int lane = tid & 31;
    const int wave = tid >> 5;
    const int half = lane >> 4;             // 0: lanes 0-15, 1: lanes 16-31
    const int l16  = lane & 15;

    float* out_mag = out;                                        // [32][513][622]
    float* out_ph  = out + (size_t)NBATCH * NMAG * NFRAMES;      // [32][1024][622]

    // B fragment base: lane holds column n = l16 (freq k0+l16), K-pair 2*half.
    const float* ldsR = &lds[l16 * LDS_STRIDE + 2 * half];
    const float* ldsI = ldsR + 16 * LDS_STRIDE;

    for (int j = 0; j < TPW; ++j) {
        const int tt = wave * TPW + j;      // wave-uniform
        if (tt >= TTILES) break;
        const int t0 = tt * 16;

        // A fragment base: lane holds row m = l16 (frame t0+l16), K-pair 2*half.
        const int tm = t0 + l16;
        const int tc = (tm < NFRAMES) ? tm : 0;      // clamp tail rows (stores guarded)
        const float* aptr = x + (size_t)b * XLEN + (size_t)tc * HOP + 2 * half;

        v8f accR = {};
        v8f accI = {};
        #pragma unroll 8
        for (int nb = 0; nb < FFT_N; nb += 4) {
            const v2f a  = *(const v2f*)(aptr + nb);   // A 16x4 f32 fragment
            const v2f br = *(const v2f*)(ldsR + nb);   // B 4x16 f32 fragment (real)
            const v2f bi = *(const v2f*)(ldsI + nb);   // B 4x16 f32 fragment (imag)
            accR = __builtin_amdgcn_wmma_f32_16x16x4_f32(
                       false, a, false, br, (short)0, accR, false, false);
            accI = __builtin_amdgcn_wmma_f32_16x16x4_f32(
                       false, a, false, bi, (short)0, accI, false, false);
        }

        // ---- Epilogue: C/D layout -> lane holds column k=k0+l16, rows q+8*half.
        const int k = k0 + l16;
        #pragma unroll
        for (int q = 0; q < 8; ++q) {
            const int t = t0 + q + 8 * half;
            if (t < NFRAMES) {
                const float rr = accR[q];
                const float ii = accI[q];
                out_ph[(size_t)b * FFT_N * NFRAMES + (size_t)k * NFRAMES + t] =
                    atan2f(ii, rr);
                if (k < NMAG) {
                    out_mag[(size_t)b * NMAG * NFRAMES + (size_t)k * NFRAMES + t] =
                        rr * rr + ii * ii;
                }
            }
        }
    }
}

extern "C" void kernel_launch(void* const* d_in, const int* in_sizes, int n_in,
                              void* d_out, int out_size, void* d_ws, size_t ws_size,
                              hipStream_t stream)
{
    (void)in_sizes; (void)n_in; (void)out_size; (void)d_ws; (void)ws_size;
    const float* x  = (const float*)d_in[0];
    const float* rw = (const float*)d_in[1];
    const float* iw = (const float*)d_in[2];
    float* out = (float*)d_out;

    const size_t shmem = (size_t)2 * 16 * LDS_STRIDE * sizeof(float); // 131584 B (<320 KB/WGP)
    static_assert(2 * 16 * LDS_STRIDE * sizeof(float) <= 320 * 1024, "LDS budget");

    // Allow >64KB dynamic LDS (WGP has 320KB on CDNA5).
    hipFuncSetAttribute((const void*)MakeSpectrogram_wmma_f32,
                        hipFuncAttributeMaxDynamicSharedMemorySize, (int)shmem);

    dim3 grid(KTILES, NBATCH, 1);
    MakeSpectrogram_wmma_f32<<<grid, 256, shmem, stream>>>(x, rw, iw, out);
}